// KanMxN_13666585935973
// MI455X (gfx1250) — compile-verified
//
#include <hip/hip_runtime.h>

// ---------------------------------------------------------------------------
// KAN layer  out[b,o] = sum_i sum_p basis(x[i,b])[p] * coeff[i,p,o]
// = GEMM  [B=4096, K=4096] x [K=4096, O=256]  with K = (i,p), basis fused.
// fp32 end-to-end -> V_WMMA_F32_16X16X4_F32 (exact dtype path).
//
// Staged K-loop: 4 input dims per LDS stage so that
//   - all 256 threads participate in basis eval (1 thread = 1 (i,row) pair)
//   - barrier count drops 4x (64 stages = 128 barriers instead of 512)
//   - 32 WMMAs per wave between barriers for latency hiding.
// ---------------------------------------------------------------------------

#define N_IN     256
#define N_OUT    256
#define N_PARAMS 16
#define DEGREE   3
#define BATCH    4096

#define M_TILE   64     // batch rows per workgroup
#define N_TILE   64     // output cols per workgroup
#define THREADS  256    // 8 wave32s
#define I_STAGE  4      // input dims processed per LDS stage

typedef __attribute__((ext_vector_type(2))) float v2f;
typedef __attribute__((ext_vector_type(8))) float v8f;

// Cubic B-spline, 16 basis funcs, uniform extended knots on [0,1].
// knots[j] = (j - 3) * h, h = 1/13.  Cox-de Boor fully unrolled; uniform
// knots => all denominators are d*h (compile-time reciprocals).
__device__ __forceinline__ void bspline16(float t, float* __restrict__ out) {
  constexpr float h = 1.0f / (float)(N_PARAMS - DEGREE);   // 1/13
  float b[N_PARAMS + DEGREE];                              // 19 deg-0 values
#pragma unroll
  for (int j = 0; j < N_PARAMS + DEGREE; ++j) {
    const float k0 = (float)(j - DEGREE) * h;
    const float k1 = (float)(j + 1 - DEGREE) * h;
    b[j] = (t >= k0 && t < k1) ? 1.0f : 0.0f;
  }
#pragma unroll
  for (int d = 1; d <= DEGREE; ++d) {
    const float rdenom = 1.0f / ((float)d * h);            // 13/d, constexpr
#pragma unroll
    for (int j = 0; j < N_PARAMS + DEGREE - d; ++j) {
      const float kj   = (float)(j - DEGREE) * h;
      const float kjd1 = (float)(j + d + 1 - DEGREE) * h;
      const float left  = (t - kj)   * rdenom;
      const float right = (kjd1 - t) * rdenom;
      b[j] = left * b[j] + right * b[j + 1];               // uses old b[j+1]
    }
  }
#pragma unroll
  for (int p = 0; p < N_PARAMS; ++p) out[p] = b[p];
}

__global__ __launch_bounds__(THREADS)
void kan_wmma_f32_kernel(const float* __restrict__ x,      // [N_IN, BATCH]
                         const float* __restrict__ coeff,  // [N_IN, P, N_OUT]
                         float* __restrict__ out) {        // [BATCH, N_OUT]
  // LDS basis tiles: 4 dims x 64 rows x 16 params, stride 17 floats
  // (17 coprime with 64 banks -> conflict-free A-fragment reads). ~17 KB.
  __shared__ float sBasis[I_STAGE][M_TILE][N_PARAMS + 1];

  const int tid  = threadIdx.x;
  const int wave = tid >> 5;
  const int lane = tid & 31;
  const int laneLo = lane & 15;
  const int half   = lane >> 4;          // 0: lanes 0-15, 1: lanes 16-31

  const int rowTile = wave & 3;          // 4 row strips of 16
  const int colGrp  = wave >> 2;         // 0/1 -> 32-col group (2 tiles)
  const int rowBase = blockIdx.x * M_TILE;
  const int colBase = blockIdx.y * N_TILE + colGrp * 32;

  // Basis-producer mapping: thread -> (iSub, row) of the 4x64 stage group.
  const int pIsub = tid >> 6;            // 0..3
  const int pRow  = tid & 63;            // 0..63

  // A-fragment: ISA 16x4 f32 layout. Lanes 0-15 hold K=k0,k0+1;
  // lanes 16-31 hold K=k0+2,k0+3.  Same K split applies to the B 4x16 frag.
  const int aRow = rowTile * 16 + laneLo;
  const int kHi  = half * 2;

  v8f acc[2] = {};                       // two 16x16 f32 output tiles

#pragma unroll 1
  for (int ii = 0; ii < N_IN; ii += I_STAGE) {
    __syncthreads();                     // previous-stage LDS reads done

    // ---- produce basis tile: every thread evaluates one (i, row) spline ---
    {
      const float t = x[(ii + pIsub) * BATCH + rowBase + pRow];
      float bp[N_PARAMS];
      bspline16(t, bp);
#pragma unroll
      for (int p = 0; p < N_PARAMS; ++p) sBasis[pIsub][pRow][p] = bp[p];
    }

    // Pull next stage's 64KB coeff slab toward L2 (global_prefetch_b8);
    // 256 threads x 256B stride covers it exactly.
    if (ii + I_STAGE < N_IN)
      __builtin_prefetch(
          coeff + (size_t)(ii + I_STAGE) * (N_PARAMS * N_OUT) + tid * 64, 0, 3);

    __syncthreads();                     // basis tile visible to all waves

    // ---- consume: 4 dims x 4 k-steps x 2 col tiles = 32 WMMAs per wave ----
#pragma unroll
    for (int iSub = 0; iSub < I_STAGE; ++iSub) {
      const float* cRow = coeff + (size_t)(ii + iSub) * (N_PARAMS * N_OUT);
#pragma unroll
      for (int kk = 0; kk < N_PARAMS; kk += 4) {
        v2f a;
        a.x = sBasis[iSub][aRow][kk + kHi];
        a.y = sBasis[iSub][aRow][kk + kHi + 1];
#pragma unroll
        for (int ct = 0; ct < 2; ++ct) {
          const int n = colBase + ct * 16 + laneLo;
          v2f bf;
          bf.x = cRow[(kk + kHi) * N_OUT + n];
          bf.y = cRow[(kk + kHi + 1) * N_OUT + n];
          acc[ct] = __builtin_amdgcn_wmma_f32_16x16x4_f32(
              /*neg_a=*/false, a, /*neg_b=*/false, bf,
              /*c_mod=*/(short)0, acc[ct], /*reuse_a=*/false, /*reuse_b=*/false);
        }
      }
    }
  }

  // D layout: VGPR g -> M = g + 8*half (within strip), N = laneLo.
  const int mBase = rowBase + rowTile * 16 + half * 8;
#pragma unroll
  for (int ct = 0; ct < 2; ++ct) {
    const int n = colBase + ct * 16 + laneLo;
#pragma unroll
    for (int g = 0; g < 8; ++g) {
      out[(size_t)(mBase + g) * N_OUT + n] = acc[ct][g];
    }
  }
}

extern "C" void kernel_launch(void* const* d_in, const int* in_sizes, int n_in,
                              void* d_out, int out_size, void* d_ws, size_t ws_size,
                              hipStream_t stream) {
  const float* x     = (const float*)d_in[0];   // [N_IN, BATCH]
  const float* coeff = (const float*)d_in[1];   // [N_IN, N_PARAMS, N_OUT]
  float* out = (float*)d_out;                   // [BATCH, N_OUT]

  dim3 grid(BATCH / M_TILE, N_OUT / N_TILE);    // (64, 4) = 256 workgroups
  kan_wmma_f32_kernel<<<grid, THREADS, 0, stream>>>(x, coeff, out);
}